// orderedLossSp_32581621908067
// MI455X (gfx1250) — compile-verified
//
#include <hip/hip_runtime.h>
#include <hip/hip_bf16.h>

// ---- problem constants (match reference) ----
#define Bq    8
#define Mq    10
#define Sq    200
#define SPAD  208          // 13 tiles of 16
#define Nq    2048
#define Lq    128
#define BETAq 0.01f
#define EPSq  1e-6f

typedef __attribute__((ext_vector_type(2))) float v2f;
typedef __attribute__((ext_vector_type(8))) float v8f;

__device__ __forceinline__ float fexp_dev(float x, float p) {
    float sg = (x > 0.0f) ? 1.0f : ((x < 0.0f) ? -1.0f : 0.0f);
    float ax = fmaxf(fabsf(x), EPSq);
    return sg * powf(ax, p);
}

// ---- zero the accumulators (must happen every launch; graph-safe) ----
__global__ void k_init(float* accum) {
    int t = threadIdx.x;
    if (t < 3) accum[t] = 0.0f;
}

// ---- precompute superquadric surface points: float4(-2x,-2y,-2z, |p|^2) ----
__global__ void k_prep_pts(const float* __restrict__ size,
                           const float* __restrict__ shape_e,
                           const float* __restrict__ deform,
                           const float* __restrict__ etas,
                           const float* __restrict__ omegas,
                           float4* __restrict__ pts) {
    int idx = blockIdx.x * blockDim.x + threadIdx.x;
    if (idx >= Bq * Mq * SPAD) return;
    int s  = idx % SPAD;
    int bm = idx / SPAD;
    float4 out;
    if (s < Sq) {
        float eta = etas[bm * Sq + s];
        float om  = omegas[bm * Sq + s];
        float e1  = shape_e[bm * 2 + 0];
        float e2  = shape_e[bm * 2 + 1];
        float a1  = size[bm * 3 + 0];
        float a2  = size[bm * 3 + 1];
        float a3  = size[bm * 3 + 2];
        float ce = cosf(eta), se = sinf(eta);
        float co = cosf(om),  so = sinf(om);
        float fce = fexp_dev(ce, e1);
        float fse = fexp_dev(se, e1);
        float fco = fexp_dev(co, e2);
        float fso = fexp_dev(so, e2);
        float px = a1 * fce * fco;
        float py = a2 * fce * fso;
        float pz = a3 * fse;
        float f0 = deform[bm * 2 + 0] * pz / a3 + 1.0f;
        float f1 = deform[bm * 2 + 1] * pz / a3 + 1.0f;
        px *= f0; py *= f1;
        float pn2 = px * px + py * py + pz * pz;
        out = make_float4(-2.0f * px, -2.0f * py, -2.0f * pz, pn2);
    } else {
        out = make_float4(0.0f, 0.0f, 0.0f, 1e30f); // pad rows never win a min
    }
    pts[idx] = out;
}

// ---- precompute rotated/translated points: float4(x,y,z, |x|^2) ----
__global__ void k_prep_xt(const float* __restrict__ ipts,
                          const float* __restrict__ trans,
                          const float* __restrict__ rot,
                          float4* __restrict__ Xt) {
    int idx = blockIdx.x * blockDim.x + threadIdx.x;
    if (idx >= Bq * Mq * Nq) return;
    int n  = idx % Nq;
    int bm = idx / Nq;
    int b  = bm / Mq;
    float qw = rot[bm * 4 + 0], qx = rot[bm * 4 + 1];
    float qy = rot[bm * 4 + 2], qz = rot[bm * 4 + 3];
    float inv = rsqrtf(qw * qw + qx * qx + qy * qy + qz * qz);
    qw *= inv; qx *= inv; qy *= inv; qz *= inv;
    float r00 = 1.0f - 2.0f * (qy * qy + qz * qz), r01 = 2.0f * (qx * qy - qw * qz), r02 = 2.0f * (qx * qz + qw * qy);
    float r10 = 2.0f * (qx * qy + qw * qz), r11 = 1.0f - 2.0f * (qx * qx + qz * qz), r12 = 2.0f * (qy * qz - qw * qx);
    float r20 = 2.0f * (qx * qz - qw * qy), r21 = 2.0f * (qy * qz + qw * qx), r22 = 1.0f - 2.0f * (qx * qx + qy * qy);
    float cx = ipts[(b * Nq + n) * 3 + 0] - trans[bm * 3 + 0];
    float cy = ipts[(b * Nq + n) * 3 + 1] - trans[bm * 3 + 1];
    float cz = ipts[(b * Nq + n) * 3 + 2] - trans[bm * 3 + 2];
    float x = r00 * cx + r01 * cy + r02 * cz;
    float y = r10 * cx + r11 * cy + r12 * cz;
    float z = r20 * cx + r21 * cy + r22 * cz;
    Xt[idx] = make_float4(x, y, z, x * x + y * y + z * z);
}

// ---- main WMMA distance + min kernel: one block per (b,m), 8 waves ----
__global__ __launch_bounds__(256) void k_sqdist(const float4* __restrict__ pts,
                                                const float4* __restrict__ Xt,
                                                float* __restrict__ V,
                                                const float* __restrict__ probs,
                                                float* __restrict__ accum) {
    __shared__ float4 ldsX[Nq];      // 32 KB (WGP has 320 KB LDS)
    __shared__ float4 ldsP[SPAD];    // 3.3 KB
    __shared__ float  ldsMinN[SPAD]; // min over all N per s-row
    __shared__ float  wsum[8];

    int bm = blockIdx.x;
    int b  = bm / Mq;
    int m  = bm % Mq;
    int tid = threadIdx.x;

    for (int i = tid; i < Nq; i += 256) ldsX[i] = Xt[bm * Nq + i];
    for (int i = tid; i < SPAD; i += 256) {
        ldsP[i] = pts[bm * SPAD + i];
        ldsMinN[i] = 3.0e38f;
    }
    __syncthreads();

    int w  = tid >> 5;       // wave 0..7 -> n-tiles [w*16, w*16+16)
    int l  = tid & 31;
    int lr = l & 15;         // row/col index within tile
    int kh = l >> 4;         // 0: K={0,1}, 1: K={2,3}  (16x4 f32 A/B layout)

    float minS[16];
    #pragma unroll
    for (int j = 0; j < 16; ++j) minS[j] = 3.0e38f;

    for (int ts = 0; ts < 13; ++ts) {
        // A tile: rows = surface points, packed (-2p | pn2 in K=3)
        float4 p4 = ldsP[ts * 16 + lr];
        v2f a;
        a.x = kh ? p4.z : p4.x;
        a.y = kh ? p4.w : p4.y;

        float minN[8];
        #pragma unroll
        for (int v = 0; v < 8; ++v) minN[v] = 3.0e38f;

        // FULLY unrolled: keeps minS[j] in static registers (no v_movrel),
        // and exposes 16 independent WMMAs per ts for pipelining.
        #pragma unroll
        for (int j = 0; j < 16; ++j) {
            int tn = w * 16 + j;
            float4 q4 = ldsX[tn * 16 + lr];
            v2f bb;
            bb.x = kh ? q4.z : q4.x;
            bb.y = kh ? 1.0f : q4.y;   // K=3 column entry = 1 -> adds pn2
            float xn2 = q4.w;

            v8f c = {0.f, 0.f, 0.f, 0.f, 0.f, 0.f, 0.f, 0.f};
            // D = A(16x4) x B(4x16) = -2 p.x + |p|^2 ; dist = D + |x|^2
            c = __builtin_amdgcn_wmma_f32_16x16x4_f32(
                    false, a, false, bb, (short)0, c, false, false);

            #pragma unroll
            for (int v = 0; v < 8; ++v) {
                float d = fmaxf(c[v] + xn2, 0.0f);
                minN[v] = fminf(minN[v], d);
                minS[j] = fminf(minS[j], d);
            }
        }
        // reduce minN across the 16 lanes of each half (same s-row set)
        #pragma unroll
        for (int v = 0; v < 8; ++v) {
            float r = minN[v];
            r = fminf(r, __shfl_xor(r, 1, 32));
            r = fminf(r, __shfl_xor(r, 2, 32));
            r = fminf(r, __shfl_xor(r, 4, 32));
            r = fminf(r, __shfl_xor(r, 8, 32));
            if (lr == 0) {
                int s = ts * 16 + v + 8 * kh;
                // nonneg floats: uint bit order == float order
                atomicMin((unsigned int*)&ldsMinN[s], __float_as_uint(r));
            }
        }
    }

    // finalize min over S: fold the two halves (rows 0-7 vs 8-15 per tile)
    #pragma unroll
    for (int j = 0; j < 16; ++j) {
        float r = minS[j];
        r = fminf(r, __shfl_xor(r, 16, 32));
        if (kh == 0) {
            int n = (w * 16 + j) * 16 + lr;
            V[(b * Nq + n) * Mq + m] = r;
        }
    }
    __syncthreads();

    // d_mp = mean_s min_n dist ; accumulate probs*d_mp / B
    float ssum = (tid < Sq) ? ldsMinN[tid] : 0.0f;
    ssum += __shfl_xor(ssum, 1, 32);
    ssum += __shfl_xor(ssum, 2, 32);
    ssum += __shfl_xor(ssum, 4, 32);
    ssum += __shfl_xor(ssum, 8, 32);
    ssum += __shfl_xor(ssum, 16, 32);
    if (l == 0) wsum[w] = ssum;
    __syncthreads();
    if (tid == 0) {
        float t = 0.0f;
        for (int i = 0; i < 8; ++i) t += wsum[i];
        float dmp = t / (float)Sq;
        atomicAdd(&accum[1], probs[b * Mq + m] * dmp / (float)Bq);
    }
}

// ---- ordered (sorted) coverage loss: one thread per (b,n) ----
__global__ void k_ordered(const float* __restrict__ V,
                          const float* __restrict__ probs,
                          float* __restrict__ accum) {
    int idx = blockIdx.x * blockDim.x + threadIdx.x;
    if (idx >= Bq * Nq) return;
    int b = idx / Nq;
    float v[Mq], p[Mq];
    #pragma unroll
    for (int m = 0; m < Mq; ++m) {
        v[m] = V[idx * Mq + m];
        p[m] = probs[b * Mq + m];
    }
    // insertion sort ascending by v (stable)
    #pragma unroll
    for (int i = 1; i < Mq; ++i) {
        float kv = v[i], kp = p[i];
        int j = i - 1;
        while (j >= 0 && v[j] > kv) { v[j + 1] = v[j]; p[j + 1] = p[j]; --j; }
        v[j + 1] = kv; p[j + 1] = kp;
    }
    float neg = 1.0f, sum = 0.0f;
    #pragma unroll
    for (int m = 0; m < Mq; ++m) {
        sum += p[m] * neg * v[m];
        neg *= (1.0f - p[m]);
    }
    atomicAdd(&accum[0], sum / (float)(Bq * Nq));
}

// ---- KLD ----
__global__ void k_kld(const float* __restrict__ mu,
                      const float* __restrict__ logvar,
                      float* __restrict__ accum) {
    int idx = blockIdx.x * blockDim.x + threadIdx.x;
    if (idx >= Bq * Lq) return;
    float lv = logvar[idx], mm = mu[idx];
    float val = 1.0f + lv - mm * mm - expf(lv);
    atomicAdd(&accum[2], -0.5f * val / (float)Bq);
}

// ---- finalize ----
__global__ void k_final(const float* __restrict__ accum, float* __restrict__ out) {
    if (threadIdx.x == 0 && blockIdx.x == 0) {
        float prims = accum[0] + accum[1];
        float kld   = accum[2];
        out[0] = prims + BETAq * kld;  // total (regs == 0)
        out[1] = prims;
        out[2] = 0.0f;
        out[3] = kld;
    }
}

extern "C" void kernel_launch(void* const* d_in, const int* in_sizes, int n_in,
                              void* d_out, int out_size, void* d_ws, size_t ws_size,
                              hipStream_t stream) {
    (void)in_sizes; (void)n_in; (void)out_size; (void)ws_size;
    const float* ipts    = (const float*)d_in[0];
    const float* trans   = (const float*)d_in[1];
    const float* rot     = (const float*)d_in[2];
    const float* size    = (const float*)d_in[3];
    const float* shape_e = (const float*)d_in[4];
    const float* deform  = (const float*)d_in[5];
    const float* probs   = (const float*)d_in[6];
    const float* etas    = (const float*)d_in[7];
    const float* omegas  = (const float*)d_in[8];
    const float* mu      = (const float*)d_in[9];
    const float* logvar  = (const float*)d_in[10];
    float* out = (float*)d_out;

    // workspace layout (floats): accum[4] | pts[B*M*208*4] | Xt[B*M*N*4] | V[B*N*M]
    float*  ws    = (float*)d_ws;
    float*  accum = ws;
    float4* pts   = (float4*)(ws + 4);
    float4* Xt    = (float4*)(ws + 4 + Bq * Mq * SPAD * 4);
    float*  V     = ws + 4 + Bq * Mq * SPAD * 4 + Bq * Mq * Nq * 4;

    k_init<<<1, 64, 0, stream>>>(accum);
    k_prep_pts<<<(Bq * Mq * SPAD + 255) / 256, 256, 0, stream>>>(size, shape_e, deform, etas, omegas, (float4*)pts);
    k_prep_xt<<<(Bq * Mq * Nq + 255) / 256, 256, 0, stream>>>(ipts, trans, rot, Xt);
    k_sqdist<<<Bq * Mq, 256, 0, stream>>>(pts, Xt, V, probs, accum);
    k_ordered<<<(Bq * Nq + 255) / 256, 256, 0, stream>>>(V, probs, accum);
    k_kld<<<(Bq * Lq + 255) / 256, 256, 0, stream>>>(mu, logvar, accum);
    k_final<<<1, 64, 0, stream>>>(accum, out);
}